// Fourier_render_patch_23476291240301
// MI455X (gfx1250) — compile-verified
//
#include <hip/hip_runtime.h>
#include <hip/hip_bf16.h>

// Fourier patch render, specialized for h=w=512, Hf=Wf=256, F=4, DIM=81.
// out[n,ch,2r+pi,2c+pj] = sum_d feat[n, ch*81+d, r, c] * basis[d, 2*pi+pj]
// Tall-skinny GEMM (K=81, N=4) driven through V_WMMA_F32_16X16X4_F32.

#define PI_F 3.14159265358979323846f

typedef __attribute__((ext_vector_type(2))) float v2f;
typedef __attribute__((ext_vector_type(8))) float v8f;

__global__ __launch_bounds__(256) void fourier_render_wmma(
    const float* __restrict__ feat, float* __restrict__ out,
    int nTiles, int nWaves)
{
  __shared__ float sBasis[84 * 4];                 // [d][parity], zero-padded d=81..83
  __shared__ __align__(16) float sOut[8][64];      // per-wave D repack scratch

  const int tid = threadIdx.x;

  // ---- build the 4 basis vectors (rel = +-0.5), exact reference ordering ----
  if (tid < 4) {
    const int par = tid;                            // par = 2*pi + pj
    const float x = ((par & 2) ? 0.5f : -0.5f) * PI_F;  // row parity -> rel_x
    const float y = ((par & 1) ? 0.5f : -0.5f) * PI_F;  // col parity -> rel_y
    float sx[4], cx[4], sy[4], cy[4];
    for (int f = 0; f < 4; ++f) {
      const float ff = (float)(f + 1);
      sx[f] = sinf(ff * x); cx[f] = cosf(ff * x);
      sy[f] = sinf(ff * y); cy[f] = cosf(ff * y);
    }
    float b[84];
    int k = 0;
    b[k++] = 1.0f;
    for (int f = 1; f <= 4; ++f) {
      const int a = f - 1;
      b[k++] = sx[a]; b[k++] = cx[a]; b[k++] = sy[a]; b[k++] = cy[a];
      for (int ft = 1; ft < f; ++ft) {
        const int t = ft - 1;
        // p_fns order: cc, cs, sc, ss; each with (fi,fti) then (fti,fi)
        b[k++] = cx[a] * cy[t]; b[k++] = cx[t] * cy[a];
        b[k++] = cx[a] * sy[t]; b[k++] = cx[t] * sy[a];
        b[k++] = sx[a] * cy[t]; b[k++] = sx[t] * cy[a];
        b[k++] = sx[a] * sy[t]; b[k++] = sx[t] * sy[a];
      }
      b[k++] = cx[a] * cy[a]; b[k++] = cx[a] * sy[a];
      b[k++] = sx[a] * cy[a]; b[k++] = sx[a] * sy[a];
    }
    b[81] = 0.0f; b[82] = 0.0f; b[83] = 0.0f;       // zero pad: K 81 -> 84
    for (int d = 0; d < 84; ++d) sBasis[d * 4 + par] = b[d];
  }
  __syncthreads();

  const int lane = tid & 31;       // wave32
  const int wave = tid >> 5;
  const int hi   = lane >> 4;      // K-half select (A & B f32 layouts)
  const int lo   = lane & 15;      // M index (A) / N index (B/C/D)

  // ---- B fragments (4x16 f32 layout: VGPR j, lane l -> K = j + 2*(l>=16)) ----
  // Held in registers, invariant over all tiles this wave processes.
  v2f Bf[21];
  #pragma unroll
  for (int s = 0; s < 21; ++s) {
    const int d0 = 4 * s + 2 * hi;
    float b0 = sBasis[d0 * 4 + (lo & 3)];
    float b1 = sBasis[(d0 + 1) * 4 + (lo & 3)];
    if (lo >= 4) { b0 = 0.0f; b1 = 0.0f; }          // only N=0..3 carry data
    Bf[s].x = b0; Bf[s].y = b1;
  }

  float* wls = &sOut[wave][0];
  const int gw = blockIdx.x * (blockDim.x >> 5) + wave;

  for (int t = gw; t < nTiles; t += nWaves) {
    const int plane = t >> 12;                       // n*3 + ch
    const int pix0  = (t & 4095) << 4;               // 16 pixels, same feature row
    const size_t planeOff = (size_t)plane * (size_t)(81u * 65536u);
    const int px = pix0 + lo;                        // A: lane lo -> pixel M

    v8f acc = {0.f, 0.f, 0.f, 0.f, 0.f, 0.f, 0.f, 0.f};

    #pragma unroll
    for (int s = 0; s < 21; ++s) {
      const int d0 = 4 * s + 2 * hi;
      const int e0 = (d0     > 80) ? 80 : d0;        // clamp pad planes (basis=0)
      const int e1 = (d0 + 1 > 80) ? 80 : (d0 + 1);
      v2f a;
      a.x = __builtin_nontemporal_load(feat + planeOff + (size_t)e0 * 65536u + px);
      a.y = __builtin_nontemporal_load(feat + planeOff + (size_t)e1 * 65536u + px);
      acc = __builtin_amdgcn_wmma_f32_16x16x4_f32(
          false, a, false, Bf[s], (short)0, acc, false, false);
    }

    // ---- repack D (only N=0..3 useful) into 2 coalesced output rows ----
    #pragma unroll
    for (int k = 0; k < 8; ++k) {
      if (lo < 4) {
        const int m  = k + 8 * hi;                   // D: VGPR k, lane-half -> M
        const int pi = lo >> 1, pj = lo & 1;
        wls[pi * 32 + 2 * m + pj] = acc[k];
      }
    }
    __asm__ volatile("s_wait_dscnt 0" ::: "memory"); // same-wave DS RAW fence
    const v2f o = *(const v2f*)(wls + hi * 32 + 2 * lo);

    const int r   = pix0 >> 8;
    const int cc0 = pix0 & 255;
    const size_t obase = ((size_t)plane * 512u + (size_t)(2 * r + hi)) * 512u
                       + (size_t)(2 * cc0 + 2 * lo);
    __builtin_nontemporal_store(o, (v2f*)(out + obase));
  }
}

extern "C" void kernel_launch(void* const* d_in, const int* in_sizes, int n_in,
                              void* d_out, int out_size, void* d_ws, size_t ws_size,
                              hipStream_t stream) {
  (void)in_sizes; (void)n_in; (void)out_size; (void)d_ws; (void)ws_size;
  const float* feat = (const float*)d_in[0];   // [4, 243, 256, 256] f32
  float* out = (float*)d_out;                  // [4, 3, 512, 512] f32

  const int nTiles = 12 * 4096;                // 12 planes x (65536/16) pixel tiles
  const int blocks = 1536;                     // 8 waves/block -> 12288 waves, 4 tiles each
  const int nWaves = blocks * 8;
  fourier_render_wmma<<<blocks, 256, 0, stream>>>(feat, out, nTiles, nWaves);
}